// DDC2Loss_68822555951365
// MI455X (gfx1250) — compile-verified
//
#include <hip/hip_runtime.h>

typedef __attribute__((ext_vector_type(2))) float v2f;
typedef __attribute__((ext_vector_type(8))) float v8f;

#define ROWS 16384
#define COLS 512
#define ROWCHUNKS 128
#define ROWS_PER_CHUNK (ROWS / ROWCHUNKS)     // 128
#define COLGROUPS (COLS / 16)                 // 32
#define WAVES_TOTAL (ROWCHUNKS * COLGROUPS)   // 4096
#define BLOCK 256
#define WAVES_PER_BLOCK (BLOCK / 32)          // 8
#define GRID (WAVES_TOTAL / WAVES_PER_BLOCK)  // 512

// Kernel 1: streaming pass. Each wave reduces a 128-row x 16-col slab.
// Column sums via V_WMMA_F32_16X16X4_F32 (A = ones 16x4, B = 4x16 data tile):
//   D[i][j] = sum_k B[k][j]  -> colsums replicated down all 16 rows of C.
// Sum of squares via per-lane fp32 FMA + wave shuffle reduction.
__global__ __launch_bounds__(BLOCK)
void ddc2_pass1_kernel(const float* __restrict__ A,
                       float* __restrict__ ws_cols,   // [ROWCHUNKS][COLS]
                       float* __restrict__ ws_q) {    // [WAVES_TOTAL]
    const int lane = threadIdx.x & 31;
    const int wave = threadIdx.x >> 5;
    const int gw   = blockIdx.x * WAVES_PER_BLOCK + wave;

    const int colgroup = gw & (COLGROUPS - 1);
    const int rowchunk = gw >> 5;               // constant within a block
    const int colbase  = colgroup * 16;

    // B-matrix (4x16 fp32, 2 VGPRs) lane mapping:
    //   VGPR0: lanes 0-15 -> K=0,N=lane ; lanes 16-31 -> K=2,N=lane-16
    //   VGPR1: lanes 0-15 -> K=1       ; lanes 16-31 -> K=3
    const int n   = lane & 15;
    const int khi = (lane >> 4) << 1;           // 0 or 2

    const float* __restrict__ base =
        A + (size_t)(rowchunk * ROWS_PER_CHUNK + khi) * COLS + colbase + n;

    v8f acc = {};                    // fp32 C accumulator (16x16)
    v2f ones; ones.x = 1.0f; ones.y = 1.0f;   // A = ones(16x4)
    float q = 0.0f;

#pragma unroll 4
    for (int r = 0; r < ROWS_PER_CHUNK; r += 4) {
        float b0 = base[(size_t)r * COLS];          // row r+khi
        float b1 = base[(size_t)r * COLS + COLS];   // row r+khi+1
        v2f b; b.x = b0; b.y = b1;
        // (neg_a, A, neg_b, B, c_mod, C, reuse_a, reuse_b)
        acc = __builtin_amdgcn_wmma_f32_16x16x4_f32(
            false, ones, false, b, (short)0, acc, false, false);
        q = __builtin_fmaf(b0, b0, q);
        q = __builtin_fmaf(b1, b1, q);
    }

    // Wave-level reduction of q across 32 lanes (wave32).
    for (int off = 16; off > 0; off >>= 1)
        q += __shfl_down(q, off, 32);
    if (lane == 0)
        ws_q[gw] = q;

    // C/D layout: element 0 = row M=0 (lanes 0-15 hold N=0..15). All rows of D
    // are identical (A is all-ones), so lanes 0-15, elem 0 carry the colsums.
    if (lane < 16)
        ws_cols[rowchunk * COLS + colbase + lane] = acc[0];
}

// Kernel 2: single block. fp64 accumulation (total and sq nearly cancel, so
// the final combine must not lose bits).
__global__ __launch_bounds__(512)
void ddc2_pass2_kernel(const float* __restrict__ ws_cols,
                       const float* __restrict__ ws_q,
                       float* __restrict__ out) {
    __shared__ double sdot[512];
    __shared__ double ssq[512];
    const int t = threadIdx.x;

    double s = 0.0;
    for (int rc = 0; rc < ROWCHUNKS; ++rc)
        s += (double)ws_cols[rc * COLS + t];

    double q = 0.0;
    for (int i = t; i < WAVES_TOTAL; i += 512)
        q += (double)ws_q[i];

    sdot[t] = s * s;
    ssq[t]  = q;
    __syncthreads();

    for (int off = 256; off > 0; off >>= 1) {
        if (t < off) {
            sdot[t] += sdot[t + off];
            ssq[t]  += ssq[t + off];
        }
        __syncthreads();
    }

    if (t == 0) {
        const double total = sdot[0];
        const double sq    = ssq[0];
        const double denom = (double)ROWS * (double)(ROWS - 1);
        out[0] = (float)((total - sq) / denom);
    }
}

extern "C" void kernel_launch(void* const* d_in, const int* in_sizes, int n_in,
                              void* d_out, int out_size, void* d_ws, size_t ws_size,
                              hipStream_t stream) {
    (void)in_sizes; (void)n_in; (void)out_size; (void)ws_size;
    const float* A = (const float*)d_in[0];
    float* out = (float*)d_out;

    float* ws_cols = (float*)d_ws;                     // 128*512 floats = 256 KiB
    float* ws_q    = ws_cols + ROWCHUNKS * COLS;       // 4096 floats = 16 KiB

    ddc2_pass1_kernel<<<GRID, BLOCK, 0, stream>>>(A, ws_cols, ws_q);
    ddc2_pass2_kernel<<<1, 512, 0, stream>>>(ws_cols, ws_q, out);
}